// GroupContrastLoss_54417235640830
// MI455X (gfx1250) — compile-verified
//
#include <hip/hip_runtime.h>
#include <math.h>

// ---------------------------------------------------------------------------
// GroupContrastLoss on gfx1250 (CDNA5) using V_WMMA_F32_16X16X4_F32.
// feat: [4,256,256,256] f32, gt: [4,19,256,256] f32 (0/1). Output: scalar f32.
// Bandwidth-bound (~0.58 GB traffic); f32 WMMA keeps full precision.
// ---------------------------------------------------------------------------

typedef __attribute__((ext_vector_type(2))) float v2f;
typedef __attribute__((ext_vector_type(8))) float v8f;

#define WMMA4(a, b, c) \
  __builtin_amdgcn_wmma_f32_16x16x4_f32(false, (a), false, (b), (short)0, (c), false, false)

static constexpr int   CB      = 4;        // batch
static constexpr int   CC      = 256;      // channels
static constexpr int   CHW     = 65536;    // H*W
static constexpr int   CK      = 19;       // classes
static constexpr int   CKP     = 32;       // classes padded to 2 WMMA N-tiles
static constexpr float CEPS    = 1e-12f;
static constexpr float CINVTAU = 1.0f / 0.07f;

static constexpr int SB = 68;   // 64-px tile, pad so LDS stride%64==4 (conflict-free WMMA reads)
static constexpr int SD = 132;  // 128-px tile, pad likewise

// ---------------------------------------------------------------------------
__global__ void zero_kernel(float* __restrict__ p, int n) {
  int i = blockIdx.x * blockDim.x + threadIdx.x;
  if (i < n) p[i] = 0.0f;
}

// ---------------------------------------------------------------------------
// k0[k][c] = sum_p pos[k][p] * fnorm[c][p]   (WMMA: A=pos^T 16x4, B=fnorm 4x16)
// Also accumulates num_pos.
// ---------------------------------------------------------------------------
__global__ void k0_kernel(const float* __restrict__ feat,
                          const float* __restrict__ gt,
                          float* __restrict__ k0raw,
                          float* __restrict__ scal,
                          int nChunks, int gridB) {
  extern __shared__ float smem[];
  float* featS = smem;                     // 256*SB
  float* posS  = featS + 256 * SB;         // 32*SB
  float* invS  = posS + 32 * SB;           // 64
  float* red   = invS + 64;                // 256

  const int t    = threadIdx.x;
  const int lane = t & 31;
  const int w    = t >> 5;                 // wave 0..7
  const int li   = lane & 15;
  const int hi   = lane >> 4;
  const int ch0  = (2 * w) * 16 + li;      // this wave's two channel tiles
  const int ch1  = (2 * w + 1) * 16 + li;

  v8f acc00 = {}, acc01 = {}, acc10 = {}, acc11 = {};
  float nposLocal = 0.0f;

  for (int chunk = blockIdx.x; chunk < nChunks; chunk += gridB) {
    const int  pix0  = chunk << 6;                 // 64 px per chunk (divides HW)
    const int  b     = pix0 >> 16;
    const int  hw    = pix0 & 65535;
    const long fbase = (long)b * CC * CHW + hw;
    const long gbase = (long)b * CK * CHW + hw;

    // ---- stage feat tile (coalesced) + per-pixel sum of squares ----
    {
      const int px = t & 63;
      const int cq = t >> 6;                       // 4 channel quarters
      float s = 0.0f;
      for (int i = 0; i < 64; ++i) {
        const int c = (cq << 6) + i;
        const float v = feat[fbase + (long)c * CHW + px];
        featS[c * SB + px] = v;
        s += v * v;
      }
      red[t] = s;
    }
    // ---- stage pos tile, classes padded to 32 (rows 19..31 = 0) ----
    for (int idx = t; idx < CKP * 64; idx += 256) {
      const int k  = idx >> 6;
      const int px = idx & 63;
      float pv = 0.0f;
      if (k < CK) {
        pv = (gt[gbase + (long)k * CHW + px] == 1.0f) ? 1.0f : 0.0f;
        nposLocal += pv;
      }
      posS[k * SB + px] = pv;
    }
    __syncthreads();
    if (t < 64) {
      const float s = red[t] + red[t + 64] + red[t + 128] + red[t + 192];
      invS[t] = 1.0f / fmaxf(sqrtf(s), CEPS);
    }
    __syncthreads();

    // ---- WMMA accumulate over the 64 pixels (K dim), 4 at a time ----
    for (int p0 = 0; p0 < 64; p0 += 4) {
      const int pA = p0 + 2 * hi;                  // A/B VGPR0 holds K=pA, VGPR1 K=pA+1
      const float ia = invS[pA];
      const float ib = invS[pA + 1];
      v2f a0, a1, b0, b1;
      a0[0] = posS[li * SB + pA];        a0[1] = posS[li * SB + pA + 1];
      a1[0] = posS[(16 + li) * SB + pA]; a1[1] = posS[(16 + li) * SB + pA + 1];
      b0[0] = featS[ch0 * SB + pA] * ia; b0[1] = featS[ch0 * SB + pA + 1] * ib;
      b1[0] = featS[ch1 * SB + pA] * ia; b1[1] = featS[ch1 * SB + pA + 1] * ib;
      acc00 = WMMA4(a0, b0, acc00);
      acc01 = WMMA4(a0, b1, acc01);
      acc10 = WMMA4(a1, b0, acc10);
      acc11 = WMMA4(a1, b1, acc11);
    }
    __syncthreads();                               // before next chunk rewrites LDS
  }

  // ---- write back: C/D layout -> class row = r + 8*hi (per tile) ----
#pragma unroll
  for (int r = 0; r < 8; ++r) {
    const int rowHi = r + 8 * hi;
    atomicAdd(&k0raw[(0  + rowHi) * CC + ch0], acc00[r]);
    atomicAdd(&k0raw[(0  + rowHi) * CC + ch1], acc01[r]);
    atomicAdd(&k0raw[(16 + rowHi) * CC + ch0], acc10[r]);
    atomicAdd(&k0raw[(16 + rowHi) * CC + ch1], acc11[r]);
  }

  red[t] = nposLocal;
  __syncthreads();
  for (int off = 128; off > 0; off >>= 1) {
    if (t < off) red[t] += red[t + off];
    __syncthreads();
  }
  if (t == 0) atomicAdd(&scal[1], red[0]);
}

// ---------------------------------------------------------------------------
// Row-normalize k0 (19 real rows), store channel-major k0nT[256][32].
// ---------------------------------------------------------------------------
__global__ void k0norm_kernel(const float* __restrict__ k0raw,
                              float* __restrict__ k0nT) {
  __shared__ float red[256];
  const int t = threadIdx.x;
  for (int k = 0; k < CKP; ++k) {
    const float v = k0raw[k * CC + t];
    red[t] = v * v;
    __syncthreads();
    for (int off = 128; off > 0; off >>= 1) {
      if (t < off) red[t] += red[t + off];
      __syncthreads();
    }
    const float inv = 1.0f / fmaxf(sqrtf(red[0]), CEPS);
    __syncthreads();                               // red[0] consumed before next k
    k0nT[t * CKP + k] = v * inv;                   // rows k>=19 are zero -> stores 0
  }
}

// ---------------------------------------------------------------------------
// logits = fnorm @ k0nT (WMMA), log-softmax over 19 classes, loss accumulate.
// ---------------------------------------------------------------------------
__global__ void loss_kernel(const float* __restrict__ feat,
                            const float* __restrict__ gt,
                            const float* __restrict__ k0nT,
                            float* __restrict__ scal) {
  extern __shared__ float smem[];
  float* featS = smem;                     // 256*SD
  float* posS  = featS + 256 * SD;         // 32*SD
  float* k0S   = posS + 32 * SD;           // 256*32
  float* invS  = k0S + CC * CKP;           // 128
  float* red   = invS + 128;               // 256

  const int t    = threadIdx.x;
  const int lane = t & 31;
  const int w    = t >> 5;
  const int li   = lane & 15;
  const int hi   = lane >> 4;

  const int  pix0  = blockIdx.x << 7;              // 128 px per block
  const int  b     = pix0 >> 16;
  const int  hw    = pix0 & 65535;
  const long fbase = (long)b * CC * CHW + hw;
  const long gbase = (long)b * CK * CHW + hw;

  for (int idx = t; idx < CC * CKP; idx += 256) k0S[idx] = k0nT[idx];

  {
    const int px = t & 127;
    const int cq = t >> 7;
    float s = 0.0f;
    for (int i = 0; i < 128; ++i) {
      const int c = (cq << 7) + i;
      const float v = feat[fbase + (long)c * CHW + px];
      featS[c * SD + px] = v;
      s += v * v;
    }
    red[t] = s;
  }
  for (int idx = t; idx < CKP * 128; idx += 256) {
    const int k  = idx >> 7;
    const int px = idx & 127;
    float pv = 0.0f;
    if (k < CK) pv = (gt[gbase + (long)k * CHW + px] == 1.0f) ? 1.0f : 0.0f;
    posS[k * SD + px] = pv;
  }
  __syncthreads();
  if (t < 128) invS[t] = 1.0f / fmaxf(sqrtf(red[t] + red[t + 128]), CEPS);
  __syncthreads();

  // ---- WMMA: each wave owns a 16-pixel subtile, both 16-class tiles ----
  const int pxb = w << 4;
  const int px  = pxb + li;                        // A: M = li for both lane halves
  const float iv = invS[px];
  v8f acc0 = {}, acc1 = {};
  for (int c0 = 0; c0 < 256; c0 += 4) {
    const int ca = c0 + 2 * hi;                    // VGPR0 K=ca, VGPR1 K=ca+1
    v2f a, b0, b1;
    a[0]  = featS[ca * SD + px] * iv;
    a[1]  = featS[(ca + 1) * SD + px] * iv;
    b0[0] = k0S[ca * CKP + li];       b0[1] = k0S[(ca + 1) * CKP + li];
    b1[0] = k0S[ca * CKP + 16 + li];  b1[1] = k0S[(ca + 1) * CKP + 16 + li];
    acc0 = WMMA4(a, b0, acc0);
    acc1 = WMMA4(a, b1, acc1);
  }

  // ---- per-pixel log-softmax: 32 logits live on one 16-lane half ----
  const bool v1 = (16 + li) < CK;                  // classes 16..18 valid
  float lacc = 0.0f;
#pragma unroll
  for (int r = 0; r < 8; ++r) {
    const int pl = pxb + r + 8 * hi;               // D layout: row = r + 8*hi
    const float l0 = acc0[r] * CINVTAU;
    const float l1 = v1 ? acc1[r] * CINVTAU : -3.0e38f;
    float mx = fmaxf(l0, l1);
    mx = fmaxf(mx, __shfl_xor(mx, 1, 32));
    mx = fmaxf(mx, __shfl_xor(mx, 2, 32));
    mx = fmaxf(mx, __shfl_xor(mx, 4, 32));
    mx = fmaxf(mx, __shfl_xor(mx, 8, 32));
    float e = __expf(l0 - mx) + (v1 ? __expf(l1 - mx) : 0.0f);
    e += __shfl_xor(e, 1, 32);
    e += __shfl_xor(e, 2, 32);
    e += __shfl_xor(e, 4, 32);
    e += __shfl_xor(e, 8, 32);
    const float logZ = mx + __logf(e);
    const float p0v = posS[li * SD + pl];
    const float p1v = v1 ? posS[(16 + li) * SD + pl] : 0.0f;
    lacc += p0v * (l0 - logZ) + p1v * (l1 - logZ);
  }

  red[t] = lacc;
  __syncthreads();
  for (int off = 128; off > 0; off >>= 1) {
    if (t < off) red[t] += red[t + off];
    __syncthreads();
  }
  if (t == 0) atomicAdd(&scal[0], red[0]);
}

// ---------------------------------------------------------------------------
__global__ void finalize_kernel(const float* __restrict__ scal,
                                float* __restrict__ out) {
  out[0] = -scal[0] / scal[1];
}

// ---------------------------------------------------------------------------
extern "C" void kernel_launch(void* const* d_in, const int* in_sizes, int n_in,
                              void* d_out, int out_size, void* d_ws, size_t ws_size,
                              hipStream_t stream) {
  (void)in_sizes; (void)n_in; (void)out_size; (void)ws_size;
  const float* feat = (const float*)d_in[0];
  const float* gt   = (const float*)d_in[1];
  float* out = (float*)d_out;
  float* ws  = (float*)d_ws;

  float* k0raw = ws;                    // 32*256
  float* k0nT  = ws + CKP * CC;         // 256*32 (channel-major)
  float* scal  = ws + 2 * CKP * CC;     // [0]=sum(pos*logp), [1]=num_pos

  const int nZero = 2 * CKP * CC + 2;
  zero_kernel<<<(nZero + 255) / 256, 256, 0, stream>>>(ws, nZero);

  const int nChunksB = (CB * CHW) / 64;                                   // 4096
  const int gridB    = 512;
  const size_t shmB  = (size_t)(256 * SB + 32 * SB + 64 + 256) * sizeof(float);
  k0_kernel<<<gridB, 256, shmB, stream>>>(feat, gt, k0raw, scal, nChunksB, gridB);

  k0norm_kernel<<<1, 256, 0, stream>>>(k0raw, k0nT);

  const int gridD   = (CB * CHW) / 128;                                   // 2048
  const size_t shmD = (size_t)(256 * SD + 32 * SD + CC * CKP + 128 + 256) * sizeof(float);
  loss_kernel<<<gridD, 256, shmD, stream>>>(feat, gt, k0nT, scal);

  finalize_kernel<<<1, 1, 0, stream>>>(scal, out);
}